// graph_23622320128649
// MI455X (gfx1250) — compile-verified
//
#include <hip/hip_runtime.h>
#include <stdint.h>

// ---------------------------------------------------------------------------
// CDNA5 async global->LDS helpers (ASYNCcnt-tracked), builtin with asm fallback
// ---------------------------------------------------------------------------
__device__ __forceinline__ void async_load_b32_to_lds(const int* gsrc, int* ldst) {
#if __has_builtin(__builtin_amdgcn_global_load_async_to_lds_b32)
    typedef __attribute__((address_space(1))) int* gptr_t;   // builtin wants non-const
    typedef __attribute__((address_space(3))) int* lptr_t;
    __builtin_amdgcn_global_load_async_to_lds_b32((gptr_t)const_cast<int*>(gsrc),
                                                  (lptr_t)ldst, 0, 0);
#else
    uint32_t loff = (uint32_t)(uintptr_t)ldst;      // low 32 bits = LDS offset
    uint64_t gaddr = (uint64_t)(uintptr_t)gsrc;
    asm volatile("global_load_async_to_lds_b32 %0, %1, off"
                 :: "v"(loff), "v"(gaddr) : "memory");
#endif
}

__device__ __forceinline__ void wait_asynccnt_0() {
#if __has_builtin(__builtin_amdgcn_s_wait_asynccnt)
    __builtin_amdgcn_s_wait_asynccnt(0);
#else
    asm volatile("s_wait_asynccnt 0" ::: "memory");
#endif
}

__device__ __forceinline__ void wait_asynccnt_2() {
#if __has_builtin(__builtin_amdgcn_s_wait_asynccnt)
    __builtin_amdgcn_s_wait_asynccnt(2);
#else
    asm volatile("s_wait_asynccnt 2" ::: "memory");
#endif
}

// ---------------------------------------------------------------------------
// Kernel 1: transpose x [C=32, N] -> xT [N, 32]; zero x1T/x2T accumulators.
// ---------------------------------------------------------------------------
__global__ void k_transpose_zero(const float* __restrict__ x,
                                 float* __restrict__ xT,
                                 float* __restrict__ x1T,
                                 float* __restrict__ x2T,
                                 int N) {
    __shared__ float tile[32][33];                 // [node_lane][channel], pad=33
    const int n0 = blockIdx.x * 32;
    const int tx = threadIdx.x & 31;
    const int ty = threadIdx.x >> 5;               // 0..7
#pragma unroll
    for (int k = 0; k < 4; ++k) {                  // channels ty, ty+8, ...
        const int c = ty + k * 8;
        const int n = n0 + tx;
        tile[tx][c] = (n < N) ? x[c * N + n] : 0.0f;   // coalesced 128B row read
    }
    __syncthreads();
#pragma unroll
    for (int k = 0; k < 4; ++k) {                  // node rows ty, ty+8, ...
        const int nn = ty + k * 8;
        const int n = n0 + nn;
        if (n < N) {
            const int idx = n * 32 + tx;           // coalesced 128B writes
            xT[idx]  = tile[nn][tx];
            x1T[idx] = 0.0f;
            x2T[idx] = 0.0f;
        }
    }
}

// ---------------------------------------------------------------------------
// Kernel 2: edge scatter with double-buffered async index staging.
// Persistent blocks stride over 256-edge tiles. One wave32 = one edge step,
// lane = channel: 128B gathers from xT[N,32] and cacheline-coalesced
// global_atomic_add_f32 scatters into x1T/x2T. Indices for the next tile are
// prefetched into the alternate LDS buffer via GLOBAL_LOAD_ASYNC_TO_LDS_B32;
// s_wait_asynccnt<=2 releases the current tile (async loads complete in-order).
// ---------------------------------------------------------------------------
__global__ void k_edge_scatter(const float* __restrict__ xT,
                               const int* __restrict__ iInd,
                               const int* __restrict__ jInd,
                               float* __restrict__ x1T,
                               float* __restrict__ x2T,
                               int E, int numTiles) {
    __shared__ int2 pairs[2][256];                 // interleaved (i,j) per edge
    const int t    = threadIdx.x;
    const int lane = t & 31;                       // channel
    const int wave = t >> 5;                       // 0..7
    const int stride = gridDim.x;

    int tile = blockIdx.x;
    int cur  = 0;

    // Prologue: stage first tile into buffer 0.
    if (tile < numTiles) {
        const int base = tile * 256;
        if (base + t < E) {
            async_load_b32_to_lds(&iInd[base + t], &pairs[0][t].x);
            async_load_b32_to_lds(&jInd[base + t], &pairs[0][t].y);
        }
    }

    while (tile < numTiles) {
        const int nextTile = tile + stride;
        if (nextTile < numTiles) {                 // prefetch next tile's indices
            const int nb = nextTile * 256;
            if (nb + t < E) {
                async_load_b32_to_lds(&iInd[nb + t], &pairs[cur ^ 1][t].x);
                async_load_b32_to_lds(&jInd[nb + t], &pairs[cur ^ 1][t].y);
            }
            wait_asynccnt_2();                     // current tile's 2 ops done
        } else {
            wait_asynccnt_0();
        }
        __syncthreads();                           // all waves' staging visible

        const int base  = tile * 256;
        const int eBase = wave * 32;
        if (base + 256 <= E) {                     // full tile: branchless hot loop
#pragma unroll 4
            for (int s = 0; s < 32; ++s) {
                const int2 ij = pairs[cur][eBase + s];   // one ds_load_b64 broadcast
                const float gi = xT[ij.x * 32 + lane];   // 128B line gathers
                const float gj = xT[ij.y * 32 + lane];
                const float g  = gi - gj;
                atomicAdd(&x1T[ij.x * 32 + lane], g);    // coalesced 128B L2 atomics
                atomicAdd(&x2T[ij.y * 32 + lane], g);
            }
        } else {                                   // tail tile only
#pragma unroll 4
            for (int s = 0; s < 32; ++s) {
                if (base + eBase + s < E) {
                    const int2 ij = pairs[cur][eBase + s];
                    const float gi = xT[ij.x * 32 + lane];
                    const float gj = xT[ij.y * 32 + lane];
                    const float g  = gi - gj;
                    atomicAdd(&x1T[ij.x * 32 + lane], g);
                    atomicAdd(&x2T[ij.y * 32 + lane], g);
                }
            }
        }
        __syncthreads();                           // buffer free for restaging
        cur ^= 1;
        tile = nextTile;
    }
}

// ---------------------------------------------------------------------------
// Kernel 3: lap = x1 - x2, ave = max(x1, x2); transpose [N,32] -> [32,N] out.
// ---------------------------------------------------------------------------
__global__ void k_finalize(const float* __restrict__ x1T,
                           const float* __restrict__ x2T,
                           float* __restrict__ lap,
                           float* __restrict__ ave,
                           int N) {
    __shared__ float tl[32][33];                   // [channel][node_lane]
    __shared__ float ta[32][33];
    const int n0 = blockIdx.x * 32;
    const int tx = threadIdx.x & 31;
    const int ty = threadIdx.x >> 5;
#pragma unroll
    for (int k = 0; k < 4; ++k) {
        const int nn = ty + k * 8;
        const int n = n0 + nn;
        float a = 0.0f, b = 0.0f;
        if (n < N) {
            a = x1T[n * 32 + tx];                  // coalesced 128B reads
            b = x2T[n * 32 + tx];
        }
        tl[tx][nn] = a - b;
        ta[tx][nn] = fmaxf(a, b);
    }
    __syncthreads();
#pragma unroll
    for (int k = 0; k < 4; ++k) {
        const int c = ty + k * 8;
        const int n = n0 + tx;
        if (n < N) {
            lap[c * N + n] = tl[c][tx];            // coalesced 128B writes
            ave[c * N + n] = ta[c][tx];
        }
    }
}

// ---------------------------------------------------------------------------
extern "C" void kernel_launch(void* const* d_in, const int* in_sizes, int n_in,
                              void* d_out, int out_size, void* d_ws, size_t ws_size,
                              hipStream_t stream) {
    const float* x    = (const float*)d_in[0];     // [1, 32, N]
    const int*   iInd = (const int*)d_in[1];       // [E]
    const int*   jInd = (const int*)d_in[2];       // [E]

    const int C = 32;
    const int N = in_sizes[0] / C;                 // 50000
    const int E = in_sizes[1];                     // 1600000

    float* lap = (float*)d_out;                    // [C, N]
    float* ave = lap + (size_t)C * N;              // [C, N]

    float* xT  = (float*)d_ws;                     // [N, C]
    float* x1T = xT  + (size_t)N * C;              // [N, C]
    float* x2T = x1T + (size_t)N * C;              // [N, C]

    const int nodeBlocks = (N + 31) / 32;
    const int numTiles   = (E + 255) / 256;
    const int scatterBlocks = numTiles < 2048 ? numTiles : 2048;

    k_transpose_zero<<<nodeBlocks, 256, 0, stream>>>(x, xT, x1T, x2T, N);
    k_edge_scatter<<<scatterBlocks, 256, 0, stream>>>(xT, iInd, jInd, x1T, x2T, E, numTiles);
    k_finalize<<<nodeBlocks, 256, 0, stream>>>(x1T, x2T, lap, ave, N);
}